// RegressionSepFNP_592705487562
// MI455X (gfx1250) — compile-verified
//
#include <hip/hip_runtime.h>
#include <math.h>

// ---------------------------------------------------------------------------
// FNP regression forward, fp32, WMMA f32 16x16x4 + async-to-LDS on gfx1250.
// ---------------------------------------------------------------------------

typedef __attribute__((ext_vector_type(2))) float v2f;
typedef __attribute__((ext_vector_type(8))) float v8f;
typedef __attribute__((ext_vector_type(4))) int   v4i;

#define AS1 __attribute__((address_space(1)))
#define AS3 __attribute__((address_space(3)))

#define F_LOG2PI 1.8378770664093453f
#define F_TEMP   0.3f
#define C_NR   2048
#define C_NM   2048
#define C_NALL 4096
#define C_DX   128
#define C_DH   128
#define C_DU   64
#define C_DZ   64

// workspace layout (float offsets)
#define OFF_SR   0u
#define OFF_U    128u
#define OFF_Z    (OFF_U    + 262144u)
#define OFF_QZM  (OFF_Z    + 262144u)
#define OFF_QZL  (OFF_QZM  + 262144u)
#define OFF_N2   (OFF_QZL  + 262144u)
#define OFF_RS   (OFF_N2   + 4096u)
#define OFF_PZM  (OFF_RS   + 4096u)
#define OFF_PZL  (OFF_PZM  + 262144u)
#define OFF_W1T  (OFF_PZL  + 262144u)
#define OFF_WUT  (OFF_W1T  + 16384u)
#define OFF_WZT  (OFF_WUT  + 16384u)
#define OFF_QZMT (OFF_WZT  + 16384u)
#define OFF_QZLT (OFF_QZMT + 131072u)
#define OFF_W    (OFF_QZLT + 131072u)

__device__ __forceinline__ v8f wmma4(v2f a, v2f b, v8f c) {
  // D = A(16x4,f32) * B(4x16,f32) + C(16x16,f32)
  return __builtin_amdgcn_wmma_f32_16x16x4_f32(false, a, false, b,
                                               (short)0, c, false, false);
}

__device__ __forceinline__ float normal_lp(float x, float mean, float logscale) {
  float t = (x - mean) * __expf(-logscale);
  return -0.5f * F_LOG2PI - logscale - 0.5f * t * t;
}

// --------------------------------------------------------------- init
__global__ void k_init(float* rowsum, float* out) {
  int t = blockIdx.x * blockDim.x + threadIdx.x;
  for (int i = t; i < C_NALL; i += gridDim.x * blockDim.x) rowsum[i] = 0.f;
  if (t == 0) { out[4096] = 0.f; out[4097] = 0.f; }
}

// --------------------------------------------------------------- transpose weights
// W1/Wu/Wz (128x128 row-major) -> col-major so B fragments are contiguous in K.
__global__ void k_transw(const float* __restrict__ W1, const float* __restrict__ Wu,
                         const float* __restrict__ Wz, float* __restrict__ W1t,
                         float* __restrict__ Wut, float* __restrict__ Wzt) {
  int e = blockIdx.x * 256 + threadIdx.x;     // 3*16384
  int mat = e >> 14, idx = e & 16383;
  int k = idx >> 7, c = idx & 127;
  const float* s = (mat == 0) ? W1 : (mat == 1) ? Wu : Wz;
  float*       d = (mat == 0) ? W1t : (mat == 1) ? Wut : Wzt;
  d[c * 128 + k] = s[k * 128 + c];
}

// qz_mean/qz_logscale rows [0,2048) -> (64 x 2048) col-major.
__global__ void k_transqz(const float* __restrict__ qzm, const float* __restrict__ qzl,
                          float* __restrict__ qzmT, float* __restrict__ qzlT) {
  int e = blockIdx.x * 256 + threadIdx.x;     // 2*131072
  int which = e >> 17, idx = e & 131071;
  int k = idx >> 6, j = idx & 63;
  if (which == 0) qzmT[j * 2048 + k] = qzm[k * 64 + j];
  else            qzlT[j * 2048 + k] = qzl[k * 64 + j];
}

// --------------------------------------------------------------- sR = mean(XR, axis=0)
__global__ void k_colmean(const float* __restrict__ XR, float* __restrict__ sR) {
  int j = threadIdx.x;                 // 128 threads
  float s = 0.f;
  for (int i = 0; i < C_NR; ++i) s += XR[i * C_DX + j];
  sR[j] = s * (1.0f / (float)C_NR);
}

// --------------------------------------------------------------- fused MLP
// block: 256 threads (8 waves), 32 rows of X_all. Each wave owns one row-tile
// and two col-tiles (c1, c1+64) sharing A fragments.
__global__ void k_mlp(const float* __restrict__ XR, const float* __restrict__ XM,
                      const float* __restrict__ W1t, const float* __restrict__ b1,
                      const float* __restrict__ Wut, const float* __restrict__ bu,
                      const float* __restrict__ Wzt, const float* __restrict__ bz,
                      const float* __restrict__ eps_u, const float* __restrict__ eps_z,
                      float* __restrict__ uo, float* __restrict__ zo,
                      float* __restrict__ qzm, float* __restrict__ qzl) {
  __shared__ float Hs[32 * 128];
  __shared__ float Ps[32 * 128];
  __shared__ float Qs[32 * 128];

  const int r0   = blockIdx.x * 32;
  const int lane = threadIdx.x & 31;
  const int wave = threadIdx.x >> 5;
  const int m    = lane & 15;
  const int kk2  = (lane >> 4) * 2;
  const int half = lane >> 4;
  const int tr   = wave & 1;
  const int c1   = (wave >> 1) * 16 + m;   // col tile 1
  const int c2   = c1 + 64;                // col tile 2

  const float* Xp = (r0 < C_NR) ? (XR + r0 * C_DX) : (XM + (r0 - C_NR) * C_DX);
  const int mbase = tr * 16 + half * 8;

  // ---- phase 1: H = relu(X @ W1 + b1)
  {
    const float* Arow = Xp + (tr * 16 + m) * C_DX + kk2;
    const float* B1 = W1t + c1 * 128 + kk2;
    const float* B2 = W1t + c2 * 128 + kk2;
    v8f acc0 = {}, acc1 = {};
    #pragma unroll 8
    for (int k = 0; k < C_DX; k += 4) {
      v2f a  = *(const v2f*)(Arow + k);
      v2f b0 = *(const v2f*)(B1 + k);
      v2f b1v = *(const v2f*)(B2 + k);
      acc0 = wmma4(a, b0, acc0);
      acc1 = wmma4(a, b1v, acc1);
    }
    const float bi1 = b1[c1], bi2 = b1[c2];
    for (int r = 0; r < 8; ++r) {
      float h0 = acc0[r] + bi1, h1 = acc1[r] + bi2;
      Hs[(mbase + r) * 128 + c1] = h0 > 0.f ? h0 : 0.f;
      Hs[(mbase + r) * 128 + c2] = h1 > 0.f ? h1 : 0.f;
    }
  }
  __syncthreads();

  // ---- phase 2: pu = H@Wu+bu, qz = H@Wz+bz (4 WMMAs share each A fragment)
  {
    const float* Arow = &Hs[(tr * 16 + m) * 128 + kk2];
    const float* BU1 = Wut + c1 * 128 + kk2;
    const float* BU2 = Wut + c2 * 128 + kk2;
    const float* BZ1 = Wzt + c1 * 128 + kk2;
    const float* BZ2 = Wzt + c2 * 128 + kk2;
    v8f aP0 = {}, aP1 = {}, aQ0 = {}, aQ1 = {};
    #pragma unroll 4
    for (int k = 0; k < C_DH; k += 4) {
      v2f a = *(const v2f*)(Arow + k);
      aP0 = wmma4(a, *(const v2f*)(BU1 + k), aP0);
      aP1 = wmma4(a, *(const v2f*)(BU2 + k), aP1);
      aQ0 = wmma4(a, *(const v2f*)(BZ1 + k), aQ0);
      aQ1 = wmma4(a, *(const v2f*)(BZ2 + k), aQ1);
    }
    const float bu1 = bu[c1], bu2 = bu[c2], bz1 = bz[c1], bz2 = bz[c2];
    for (int r = 0; r < 8; ++r) {
      Ps[(mbase + r) * 128 + c1] = aP0[r] + bu1;
      Ps[(mbase + r) * 128 + c2] = aP1[r] + bu2;
      Qs[(mbase + r) * 128 + c1] = aQ0[r] + bz1;
      Qs[(mbase + r) * 128 + c2] = aQ1[r] + bz2;
    }
  }
  __syncthreads();

  // ---- phase 3: reparameterize
  for (int e = threadIdx.x; e < 32 * 64; e += 256) {
    const int r = e >> 6, j = e & 63;
    const int g = r0 + r;
    float um = Ps[r * 128 + j], ul = Ps[r * 128 + 64 + j];
    uo[g * 64 + j] = um + __expf(ul) * eps_u[g * 64 + j];
    float qm = Qs[r * 128 + j], ql = Qs[r * 128 + 64 + j];
    qzm[g * 64 + j] = qm;
    qzl[g * 64 + j] = ql;
    zo[g * 64 + j]  = qm + __expf(ql) * eps_z[g * 64 + j];
  }
}

// --------------------------------------------------------------- |u_i|^2
__global__ void k_norms(const float* __restrict__ u, float* __restrict__ n2) {
  int i = blockIdx.x * blockDim.x + threadIdx.x;   // 4096 threads
  if (i < C_NALL) {
    float s = 0.f;
    for (int j = 0; j < C_DU; ++j) { float v = u[i * C_DU + j]; s += v * v; }
    n2[i] = s;
  }
}

// --------------------------------------------------------------- graph kernel
// block: 256 threads = 8 waves, 64x64 region of Wraw(4096x2048).
// dot = u[i].u[j] via WMMA (K=64), then relaxed-Bernoulli + rowsums.
__global__ void k_pair(const float* __restrict__ u, const float* __restrict__ n2,
                       const float* __restrict__ U_G, const float* __restrict__ U_A,
                       const float* __restrict__ g_logscale,
                       float* __restrict__ Wraw, float* __restrict__ rowsum) {
  const int C0 = blockIdx.x * 64;
  const int R0 = blockIdx.y * 64;
  const int lane = threadIdx.x & 31;
  const int wave = threadIdx.x >> 5;
  const int m = lane & 15;
  const int kk2 = (lane >> 4) * 2;
  const int half = lane >> 4;
  const int tr = wave & 3;
  const int tc1 = wave >> 2;                 // 0..1 ; second tile at tc1+2
  const float inv_gs = __expf(-g_logscale[0]);

  const float* Arow = u + (R0 + tr * 16 + m) * C_DU + kk2;
  const float* B1 = u + (C0 + tc1 * 16 + m) * C_DU + kk2;
  const float* B2 = B1 + 32 * C_DU;
  v8f acc0 = {}, acc1 = {};
  #pragma unroll
  for (int k = 0; k < C_DU; k += 4) {
    v2f a = *(const v2f*)(Arow + k);
    acc0 = wmma4(a, *(const v2f*)(B1 + k), acc0);
    acc1 = wmma4(a, *(const v2f*)(B2 + k), acc1);
  }

  const int ib = R0 + tr * 16 + half * 8;
  for (int tile = 0; tile < 2; ++tile) {
    const int j = C0 + (tc1 + 2 * tile) * 16 + m;
    const float bn = n2[j];
    for (int r = 0; r < 8; ++r) {
      const int i = ib + r;
      float dot = (tile == 0) ? acc0[r] : acc1[r];
      float d2 = fmaxf(n2[i] + bn - 2.f * dot, 0.f);
      float logp = fminf(-0.5f * d2 * inv_gs, -1e-7f);
      float logit = logp - __logf(-expm1f(logp));
      float Uv = (i < C_NR) ? U_G[i * C_NR + j] : U_A[(i - C_NR) * C_NR + j];
      Uv = fminf(fmaxf(Uv, 1e-6f), 1.f - 1e-6f);
      float x = (logit + __logf(Uv) - log1pf(-Uv)) * (1.0f / F_TEMP);
      float val = 1.f / (1.f + __expf(-x));
      Wraw[(size_t)i * C_NR + j] = val;
      float s = val;                          // reduce 16 lanes of this half
      s += __shfl_xor(s, 1, 16);
      s += __shfl_xor(s, 2, 16);
      s += __shfl_xor(s, 4, 16);
      s += __shfl_xor(s, 8, 16);
      if ((lane & 15) == 0) atomicAdd(&rowsum[i], s);
    }
  }
}

// --------------------------------------------------------------- pz = (Wraw @ [qzm|qzl]) / rowsum
// 32 rows x 128 cols, K = 2048. A staged via async-to-LDS, double buffered.
__global__ void k_pz(const float* __restrict__ Wraw,
                     const float* __restrict__ qzmT, const float* __restrict__ qzlT,
                     const float* __restrict__ rowsum,
                     float* __restrict__ pzm, float* __restrict__ pzl) {
  __shared__ float As[2][32 * 128];           // 32 KB double buffer
  const int r0 = blockIdx.x * 32;
  const int tid = threadIdx.x;
  const int lane = tid & 31;
  const int wave = tid >> 5;
  const int m = lane & 15;
  const int kk2 = (lane >> 4) * 2;
  const int half = lane >> 4;
  const int tr = wave & 1;
  const int c1 = (wave >> 1) * 16 + m;        // 0..63: tile1 -> pzm, tile2 -> pzl

  // async stage of one 32x128 K-chunk (4 x b128 per thread)
  auto stage = [&](int ch, int buf) {
    const int k0 = ch * 128;
    #pragma unroll
    for (int i = 0; i < 4; ++i) {
      const int q = tid + i * 256;            // 1024 x 16B
      const int row = q >> 5;
      const int colf = (q & 31) * 4;
      const float* gp = Wraw + (size_t)(r0 + row) * C_NR + k0 + colf;
      float* lp = &As[buf][q * 4];
      __builtin_amdgcn_global_load_async_to_lds_b128(
          (AS1 v4i*)gp, (AS3 v4i*)lp, 0, 0);
    }
  };

  v8f acc0 = {}, acc1 = {};
  stage(0, 0);
  for (int ch = 0; ch < 16; ++ch) {
    const int buf = ch & 1;
    if (ch + 1 < 16) {
      stage(ch + 1, buf ^ 1);
      __builtin_amdgcn_s_wait_asynccnt(4);    // this buffer's 4 loads done
    } else {
      __builtin_amdgcn_s_wait_asynccnt(0);
    }
    __syncthreads();

    const float* Arow = &As[buf][(tr * 16 + m) * 128 + kk2];
    const float* B1 = qzmT + (size_t)c1 * C_NR + ch * 128 + kk2;
    const float* B2 = qzlT + (size_t)c1 * C_NR + ch * 128 + kk2;
    #pragma unroll 8
    for (int k = 0; k < 128; k += 4) {
      v2f a = *(const v2f*)(Arow + k);
      acc0 = wmma4(a, *(const v2f*)(B1 + k), acc0);
      acc1 = wmma4(a, *(const v2f*)(B2 + k), acc1);
    }
    __syncthreads();
  }

  const int ib = r0 + tr * 16 + half * 8;
  for (int r = 0; r < 8; ++r) {
    const int i = ib + r;
    const float inv = 1.f / (rowsum[i] + 1e-8f);
    pzm[i * C_DZ + c1] = acc0[r] * inv;
    pzl[i * C_DZ + c1] = acc1[r] * inv;
  }
}

// --------------------------------------------------------------- log_pqz_M reduction
__global__ void k_pqz(const float* __restrict__ z,
                      const float* __restrict__ pzm, const float* __restrict__ pzl,
                      const float* __restrict__ qzm, const float* __restrict__ qzl,
                      float* __restrict__ out) {
  __shared__ float red[256];
  const int e = blockIdx.x * 256 + threadIdx.x;   // 2048*64 elements, grid 512
  const int i = C_NR + (e >> 6);
  const int j = e & 63;
  const int idx = i * C_DZ + j;
  float zv = z[idx];
  float d = normal_lp(zv, pzm[idx], pzl[idx]) - normal_lp(zv, qzm[idx], qzl[idx]);
  red[threadIdx.x] = d;
  __syncthreads();
  for (int s = 128; s > 0; s >>= 1) {
    if (threadIdx.x < s) red[threadIdx.x] += red[threadIdx.x + s];
    __syncthreads();
  }
  if (threadIdx.x == 0) atomicAdd(&out[4097], red[0]);
}

// --------------------------------------------------------------- per-row head
// block = one row n, 128 threads; streams w0 (268 MB): the HBM-bound stage.
__global__ void k_head(const float* __restrict__ sR, const float* __restrict__ z,
                       const float* __restrict__ u, const float* __restrict__ w0,
                       const float* __restrict__ w2, const float* __restrict__ yM,
                       float* __restrict__ out) {
  __shared__ float rep[256];
  __shared__ float red0[128];
  __shared__ float red1[128];
  const int n = blockIdx.x;
  const int k = threadIdx.x;

  for (int j = k; j < 256; j += 128) {
    float v;
    if      (j < 128) v = sR[j];
    else if (j < 192) v = z[(C_NR + n) * C_DZ + (j - 128)];
    else              v = u[(C_NR + n) * C_DU + (j - 192)];
    rep[j] = v;
  }
  __syncthreads();

  const float* w0p = w0 + (size_t)n * 256 * 128 + k;
  float acc = 0.f;
  #pragma unroll 4
  for (int j = 0; j < 256; ++j) {
    if (j + 16 < 256) __builtin_prefetch(w0p + (size_t)(j + 16) * 128, 0, 0);
    acc = fmaf(rep[j], w0p[(size_t)j * 128], acc);
  }
  float h1 = acc > 0.f ? acc : 0.f;

  const float* w2p = w2 + ((size_t)n * 128 + k) * 2;
  red0[k] = h1 * w2p[0];
  red1[k] = h1 * w2p[1];
  __syncthreads();
  for (int s = 64; s > 0; s >>= 1) {
    if (k < s) { red0[k] += red0[k + s]; red1[k] += red1[k + s]; }
    __syncthreads();
  }
  if (k == 0) {
    float mean_y = red0[0];
    float o1 = red1[0];
    float sp = (o1 > 0.f) ? (o1 + log1pf(__expf(-o1))) : log1pf(__expf(o1));
    float logstd = __logf(0.1f + 0.9f * sp);
    out[n] = mean_y;
    out[2048 + n] = logstd;
    atomicAdd(&out[4096], normal_lp(yM[n], mean_y, logstd));
  }
}

// ---------------------------------------------------------------------------
extern "C" void kernel_launch(void* const* d_in, const int* in_sizes, int n_in,
                              void* d_out, int out_size, void* d_ws, size_t ws_size,
                              hipStream_t stream) {
  (void)in_sizes; (void)n_in; (void)out_size; (void)ws_size;
  const float* XR    = (const float*)d_in[0];
  const float* XM    = (const float*)d_in[1];
  const float* yM    = (const float*)d_in[2];
  const float* eps_u = (const float*)d_in[3];
  const float* eps_z = (const float*)d_in[4];
  const float* U_G   = (const float*)d_in[5];
  const float* U_A   = (const float*)d_in[6];
  const float* W1    = (const float*)d_in[7];
  const float* b1    = (const float*)d_in[8];
  const float* Wu    = (const float*)d_in[9];
  const float* bu    = (const float*)d_in[10];
  const float* Wz    = (const float*)d_in[11];
  const float* bz    = (const float*)d_in[12];
  const float* gls   = (const float*)d_in[13];
  const float* w0    = (const float*)d_in[14];
  const float* w2    = (const float*)d_in[15];

  float* ws  = (float*)d_ws;
  float* sR   = ws + OFF_SR;
  float* u    = ws + OFF_U;
  float* z    = ws + OFF_Z;
  float* qzm  = ws + OFF_QZM;
  float* qzl  = ws + OFF_QZL;
  float* n2   = ws + OFF_N2;
  float* rs   = ws + OFF_RS;
  float* pzm  = ws + OFF_PZM;
  float* pzl  = ws + OFF_PZL;
  float* W1t  = ws + OFF_W1T;
  float* Wut  = ws + OFF_WUT;
  float* Wzt  = ws + OFF_WZT;
  float* qzmT = ws + OFF_QZMT;
  float* qzlT = ws + OFF_QZLT;
  float* Wraw = ws + OFF_W;
  float* out  = (float*)d_out;

  k_init    <<<16, 256, 0, stream>>>(rs, out);
  k_transw  <<<192, 256, 0, stream>>>(W1, Wu, Wz, W1t, Wut, Wzt);
  k_colmean <<<1, 128, 0, stream>>>(XR, sR);
  k_mlp     <<<C_NALL / 32, 256, 0, stream>>>(XR, XM, W1t, b1, Wut, bu, Wzt, bz,
                                              eps_u, eps_z, u, z, qzm, qzl);
  k_norms   <<<C_NALL / 128, 128, 0, stream>>>(u, n2);
  k_transqz <<<1024, 256, 0, stream>>>(qzm, qzl, qzmT, qzlT);
  k_pair    <<<dim3(C_NR / 64, C_NALL / 64), 256, 0, stream>>>(u, n2, U_G, U_A,
                                                               gls, Wraw, rs);
  k_pz      <<<C_NALL / 32, 256, 0, stream>>>(Wraw, qzmT, qzlT, rs, pzm, pzl);
  k_pqz     <<<(C_NM * C_DZ) / 256, 256, 0, stream>>>(z, pzm, pzl, qzm, qzl, out);
  k_head    <<<C_NM, 128, 0, stream>>>(sR, z, u, w0, w2, yM, out);
}